// EMCaps_56538949484598
// MI455X (gfx1250) — compile-verified
//
#include <hip/hip_runtime.h>
#include <math.h>

// ---------------------------------------------------------------------------
// EM-capsule network for MI455X (gfx1250, wave32).
// Key design decisions (reasoned from MI455X specs, no runtime available):
//  * Votes (231MB/stage if materialized) are NEVER written to HBM: each block
//    fuses vote-GEMM + fuzzy routing, votes streamed through LDS/registers.
//    320KB LDS per WGP makes the whole routing state LDS-resident.
//  * Vote matmuls use V_WMMA_F32_16X16X4_F32 (full fp32, K=4 == capsule P):
//    D[(c,r),(n,p)] = sum_q W[i,c,q,r] * pp[n,i,p,q]; B packs 4 samples x 4
//    pose rows = 16 fully used columns; A covers 16 (c,r) rows per M-tile.
//  * Primary-caps 1x1 conv = fp32 WMMA GEMM (M=16384, K=64, N=128).
// ---------------------------------------------------------------------------

typedef __attribute__((ext_vector_type(2))) float v2f;
typedef __attribute__((ext_vector_type(8))) float v8f;

#define EPSf 1e-8f
#define LAMf 1e-3f
#define G 4          // samples per caps block (fills WMMA B columns: 4n x 4p)

__device__ inline v8f wmma4(v2f a, v2f b, v8f c) {
  // D(16x16,f32) += A(16x4,f32) * B(4x16,f32)
  return __builtin_amdgcn_wmma_f32_16x16x4_f32(
      /*neg_a=*/false, a, /*neg_b=*/false, b,
      /*c_mod=*/(short)0, c, /*reuse_a=*/false, /*reuse_b=*/false);
}

// ------------------------- conv1 + BN + ReLU -------------------------------
// x:(64,1,32,32) w:(64,1,5,5) stride2 pad2 -> y:(64,64,16,16) NCHW
__global__ void conv1_bn_relu(const float* __restrict__ x,
                              const float* __restrict__ w,
                              const float* __restrict__ bias,
                              const float* __restrict__ gam,
                              const float* __restrict__ bet,
                              const float* __restrict__ mean,
                              const float* __restrict__ var,
                              float* __restrict__ y) {
  int idx = blockIdx.x * blockDim.x + threadIdx.x;
  if (idx >= 64 * 64 * 256) return;
  int ow = idx & 15, oh = (idx >> 4) & 15, oc = (idx >> 8) & 63, b = idx >> 14;
  float acc = bias[oc];
  for (int kh = 0; kh < 5; kh++) {
    int ih = oh * 2 - 2 + kh;
    if (ih < 0 || ih >= 32) continue;
    for (int kw = 0; kw < 5; kw++) {
      int iw = ow * 2 - 2 + kw;
      if (iw < 0 || iw >= 32) continue;
      acc += x[b * 1024 + ih * 32 + iw] * w[oc * 25 + kh * 5 + kw];
    }
  }
  float inv_std = rsqrtf(var[oc] + 1e-3f);
  acc = (acc - mean[oc]) * (gam[oc] * inv_std) + bet[oc];
  y[idx] = acc > 0.f ? acc : 0.f;
}

// ------------------- primary caps pose: 1x1 conv as WMMA GEMM --------------
// A[m=(b,hw)][k=ch] = y[b][k][hw], B[k][n] = w[n][k], D -> pose0[m][n], N=128
__global__ __launch_bounds__(128) void prim_pose_gemm(
    const float* __restrict__ y, const float* __restrict__ w,
    const float* __restrict__ bias, float* __restrict__ pose0) {
  int wv = threadIdx.x >> 5, lane = threadIdx.x & 31;
  int col = lane & 15, hi = lane >> 4;
  int tile = blockIdx.x * 4 + wv;       // 1024 M-tiles * 8 N-tiles = 8192
  int mt = tile >> 3, nt = tile & 7;
  int m = mt * 16 + col;                // A row for this lane
  int b = m >> 8, hw = m & 255;
  int n = nt * 16 + col;                // B col for this lane
  v8f acc = {0.f, 0.f, 0.f, 0.f, 0.f, 0.f, 0.f, 0.f};
  for (int kk = 0; kk < 16; kk++) {     // K = 64 in steps of 4
    int k0 = kk * 4 + hi * 2;
    v2f av, bv;
    av.x = y[(b * 64 + k0) * 256 + hw];
    av.y = y[(b * 64 + k0 + 1) * 256 + hw];
    bv.x = w[n * 64 + k0];
    bv.y = w[n * 64 + k0 + 1];
    acc = wmma4(av, bv, acc);
  }
  float bs = bias[n];
#pragma unroll
  for (int j = 0; j < 8; j++) {
    int row = mt * 16 + j + hi * 8;     // D row layout: VGPR j -> M=j / j+8
    pose0[row * 128 + nt * 16 + col] = acc[j] + bs;
  }
}

// ------------------- primary caps activation (tiny, scalar) ----------------
__global__ void prim_a_kernel(const float* __restrict__ y,
                              const float* __restrict__ w,
                              const float* __restrict__ bias,
                              float* __restrict__ a0) {
  int t = blockIdx.x * blockDim.x + threadIdx.x;
  if (t >= 16384 * 8) return;
  int d = t & 7, m = t >> 3;
  int b = m >> 8, hw = m & 255;
  float acc = bias[d];
  for (int k = 0; k < 64; k++) acc += y[(b * 64 + k) * 256 + hw] * w[d * 64 + k];
  a0[t] = 1.0f / (1.0f + expf(-acc));
}

// ------------------------- fused caps layer --------------------------------
// One block = 4 samples.  Wave wv owns output-capsule M-tile wv (rows (c,r)).
// Votes recomputed by WMMA in 4 streamed passes; routing state in LDS.
__device__ inline v8f caps_vote(const float* __restrict__ Wm,
                                const float* __restrict__ ppS, int i, int widx,
                                int wv, int Cout, int N, int col, int hi) {
  // A fragment: row16 = col -> (c,r); K = q = hi*2 + {0,1}
  int c = wv * 4 + (col >> 2);
  int r = col & 3;
  int q = hi * 2;
  float a0 = 0.f, a1 = 0.f;
  if (c < Cout) {
    const float* wp = Wm + (widx * Cout + c) * 16 + r;  // W[i][c][q][r]
    a0 = wp[q * 4];
    a1 = wp[(q + 1) * 4];
  }
  // B fragment: col16 = (g,p); B[q][(g,p)] = pp[g][i][p*4+q]
  const float* bp = ppS + (((col >> 2) * N) + i) * 16 + (col & 3) * 4 + q;
  v2f av, bv;
  av.x = a0; av.y = a1;
  bv.x = bp[0]; bv.y = bp[1];
  v8f z = {0.f, 0.f, 0.f, 0.f, 0.f, 0.f, 0.f, 0.f};
  return wmma4(av, bv, z);
}

__device__ inline void class_adjust(v8f& d, int i, int Hin, int Win, int Bin,
                                    int col) {
  // v[...,0] += h/H (row r=0,p=0), v[...,1] += w/W (row r=1,p=0)
  if ((col & 3) == 0) {
    float ch = (float)(i / (Win * Bin)) / (float)Hin;
    float cw = (float)((i / Bin) % Win) / (float)Win;
    d[0] += ch; d[4] += ch;   // rows with row%4==0
    d[1] += cw; d[5] += cw;   // rows with row%4==1
  }
}

__global__ __launch_bounds__(128) void caps_layer_kernel(
    const float* __restrict__ pose_in, const float* __restrict__ a_in,
    const float* __restrict__ Wm, const float* __restrict__ beta_u,
    const float* __restrict__ beta_a, float* __restrict__ pose_out,
    float* __restrict__ a_out, int Hin, int Win, int Bin, int Kk, int stride,
    int Cout, int OHW, int OW, int N, int MT, int is_class) {
  const int tid = threadIdx.x;
  const int wv = tid >> 5, lane = tid & 31, col = lane & 15, hi = lane >> 4;
  const int n0 = blockIdx.x * G;
  const int gD = col >> 2;              // sample index of this lane's D column

  extern __shared__ float smem[];
  float* ppS = smem;                    // [G][N][16] pose patches
  float* awS = ppS + G * N * 16;        // [G][N]     activations
  float* c0S = awS + G * N;             // [G][N]     coeff at t=0 (c-indep)
  float* rS  = c0S + G * N;             // [G][N][C]  routing r
  float* cfS = rS + G * N * Cout;       // [G][N][C]  coeff at t=1
  float* muS = cfS + G * N * Cout;      // [MT*16][16] mu tiles
  float* sgS = muS + MT * 256;          // [MT*16][16] sigma tiles
  float* dnS = sgS + MT * 256;          // [G][C]      sum rm + eps
  float* wsS = dnS + G * Cout;          // [4][256]    per-wave vote scratch
  float* smS = wsS + 4 * 256;           // [G]         sum of aw

  // ---- gather patches + activations into LDS ----
  for (int t = tid; t < G * N; t += 128) {
    int g = t / N, i = t - g * N;
    int b, ih, iw, d;
    if (is_class) {
      b = n0 + g;
      ih = i / (Win * Bin);
      iw = (i / Bin) % Win;
      d = i % Bin;
    } else {
      int n = n0 + g;
      b = n / OHW;
      int rem = n - b * OHW;
      int oh = rem / OW, ow = rem - oh * OW;
      int kh = i / (Kk * Bin), kw = (i / Bin) % Kk;
      d = i % Bin;
      ih = oh * stride + kh;
      iw = ow * stride + kw;
    }
    int src = ((b * Hin + ih) * Win + iw) * Bin + d;
    awS[t] = a_in[src];
    const float* pp = pose_in + (size_t)src * 16;
    float* dst = ppS + t * 16;
    for (int e = 0; e < 16; e++) dst[e] = pp[e];
  }
  __syncthreads();

  // ---- t=0 coefficients: r uniform -> coeff0 = aw / (sum aw + C^2*eps) ----
  if (tid < G) {
    float s = 0.f;
    for (int i = 0; i < N; i++) s += awS[tid * N + i];
    smS[tid] = s + (float)(Cout * Cout) * EPSf;
  }
  __syncthreads();
  for (int t = tid; t < G * N; t += 128) c0S[t] = awS[t] / smS[t / N];
  __syncthreads();

  // ---- PASS 1: mu0 = sum_i coeff0 * v ----
  if (wv < MT) {
    v8f acc = {0.f, 0.f, 0.f, 0.f, 0.f, 0.f, 0.f, 0.f};
    for (int i = 0; i < N; i++) {
      int widx = is_class ? (i % Bin) : i;
      v8f d = caps_vote(Wm, ppS, i, widx, wv, Cout, N, col, hi);
      if (is_class) class_adjust(d, i, Hin, Win, Bin, col);
      float cf = c0S[gD * N + i];
#pragma unroll
      for (int j = 0; j < 8; j++) acc[j] += cf * d[j];
    }
#pragma unroll
    for (int j = 0; j < 8; j++) muS[(wv * 16 + j + hi * 8) * 16 + col] = acc[j];
  }
  __syncthreads();

  // ---- PASS 2: dist -> inverse -> r (update for t=1) ----
  if (wv < MT) {
    float* myws = wsS + wv * 256;
    for (int i = 0; i < N; i++) {
      int widx = is_class ? (i % Bin) : i;
      v8f d = caps_vote(Wm, ppS, i, widx, wv, Cout, N, col, hi);
      if (is_class) class_adjust(d, i, Hin, Win, Bin, col);
#pragma unroll
      for (int j = 0; j < 8; j++) myws[(j + hi * 8) * 16 + col] = d[j];
      // same-wave LDS ops are in-order: safe to read staged tile now
      if (lane < 16) {
        int gl = lane >> 2, cl = lane & 3;
        int cg = wv * 4 + cl;
        if (cg < Cout) {
          float dist = EPSf;
          for (int r = 0; r < 4; r++)
            for (int p = 0; p < 4; p++) {
              float df = myws[(cl * 4 + r) * 16 + gl * 4 + p] -
                         muS[(wv * 16 + cl * 4 + r) * 16 + gl * 4 + p];
              dist += df * df;
            }
          rS[(gl * N + i) * Cout + cg] = 1.0f / dist;  // MF=2 -> dist^-1
        }
      }
    }
  }
  __syncthreads();
  // normalize r over c
  for (int t = tid; t < G * N; t += 128) {
    float* rp = rS + t * Cout;
    float s = 0.f;
    for (int c = 0; c < Cout; c++) s += rp[c];
    float inv = 1.0f / (s + EPSf);
    for (int c = 0; c < Cout; c++) rp[c] *= inv;
  }
  __syncthreads();

  // ---- coeff1 = r^2*aw / (sum_i r^2*aw + eps); denom kept for r_sum ----
  for (int t = tid; t < G * Cout; t += 128) {
    int g = t / Cout, c = t - g * Cout;
    float s = EPSf;
    for (int i = 0; i < N; i++) {
      float r = rS[(g * N + i) * Cout + c];
      s += r * r * awS[g * N + i];
    }
    dnS[t] = s;
  }
  __syncthreads();
  for (int t = tid; t < G * N * Cout; t += 128) {
    int c = t % Cout, gi = t / Cout, g = gi / N;
    float r = rS[gi * Cout + c];
    cfS[t] = r * r * awS[gi] / dnS[g * Cout + c];
  }
  __syncthreads();

  // per-lane output-capsule indices for its two D row groups
  const int cA = wv * 4 + hi * 2, cB = cA + 1;

  // ---- PASS 3: mu1 = sum_i coeff1 * v ----
  if (wv < MT) {
    v8f acc = {0.f, 0.f, 0.f, 0.f, 0.f, 0.f, 0.f, 0.f};
    for (int i = 0; i < N; i++) {
      int widx = is_class ? (i % Bin) : i;
      v8f d = caps_vote(Wm, ppS, i, widx, wv, Cout, N, col, hi);
      if (is_class) class_adjust(d, i, Hin, Win, Bin, col);
      float cfA2 = (cA < Cout) ? cfS[(gD * N + i) * Cout + cA] : 0.f;
      float cfB2 = (cB < Cout) ? cfS[(gD * N + i) * Cout + cB] : 0.f;
#pragma unroll
      for (int j = 0; j < 8; j++) acc[j] += (j < 4 ? cfA2 : cfB2) * d[j];
    }
#pragma unroll
    for (int j = 0; j < 8; j++) muS[(wv * 16 + j + hi * 8) * 16 + col] = acc[j];
  }
  __syncthreads();

  // ---- PASS 4: sigma^2 = sum_i coeff1 * (v - mu1)^2 ----
  if (wv < MT) {
    float mreg[8];
#pragma unroll
    for (int j = 0; j < 8; j++) mreg[j] = muS[(wv * 16 + j + hi * 8) * 16 + col];
    v8f acc = {0.f, 0.f, 0.f, 0.f, 0.f, 0.f, 0.f, 0.f};
    for (int i = 0; i < N; i++) {
      int widx = is_class ? (i % Bin) : i;
      v8f d = caps_vote(Wm, ppS, i, widx, wv, Cout, N, col, hi);
      if (is_class) class_adjust(d, i, Hin, Win, Bin, col);
      float cfA2 = (cA < Cout) ? cfS[(gD * N + i) * Cout + cA] : 0.f;
      float cfB2 = (cB < Cout) ? cfS[(gD * N + i) * Cout + cB] : 0.f;
#pragma unroll
      for (int j = 0; j < 8; j++) {
        float df = d[j] - mreg[j];
        acc[j] += (j < 4 ? cfA2 : cfB2) * df * df;
      }
    }
#pragma unroll
    for (int j = 0; j < 8; j++) sgS[(wv * 16 + j + hi * 8) * 16 + col] = acc[j];
  }
  __syncthreads();

  // ---- activations + pose output ----
  for (int t = tid; t < G * Cout; t += 128) {
    int g = t / Cout, c = t - g * Cout;
    float rsum = dnS[t] - EPSf;  // sum_i r^2*aw
    float slog = 0.f;
    for (int p = 0; p < 4; p++)
      for (int r = 0; r < 4; r++)
        slog += logf(sgS[(c * 4 + r) * 16 + g * 4 + p] + EPSf);
    float cost = rsum * (16.f * beta_u[c] + 0.5f * slog);
    a_out[(n0 + g) * Cout + c] = 1.0f / (1.0f + expf(-LAMf * (beta_a[c] - cost)));
  }
  if (pose_out) {
    for (int t = tid; t < G * Cout * 16; t += 128) {
      int pr = t & 15, gc = t >> 4;
      int g = gc / Cout, c = gc - g * Cout;
      int p = pr >> 2, r = pr & 3;
      pose_out[(size_t)(n0 + g) * Cout * 16 + c * 16 + pr] =
          muS[(c * 4 + r) * 16 + g * 4 + p];
    }
  }
}

// ---------------------------------------------------------------------------
static size_t caps_lds_bytes(int N, int C, int MT) {
  size_t f = (size_t)G * N * 16 + 2u * G * N + 2u * (size_t)G * N * C +
             2u * (size_t)MT * 256 + (size_t)G * C + 4 * 256 + 16;
  return f * sizeof(float);
}

extern "C" void kernel_launch(void* const* d_in, const int* in_sizes, int n_in,
                              void* d_out, int out_size, void* d_ws,
                              size_t ws_size, hipStream_t stream) {
  (void)in_sizes; (void)n_in; (void)out_size; (void)ws_size;
  const float* x   = (const float*)d_in[0];
  const float* c1w = (const float*)d_in[1];
  const float* c1b = (const float*)d_in[2];
  const float* bng = (const float*)d_in[3];
  const float* bnb = (const float*)d_in[4];
  const float* bnm = (const float*)d_in[5];
  const float* bnv = (const float*)d_in[6];
  const float* ppw = (const float*)d_in[7];
  const float* ppb = (const float*)d_in[8];
  const float* paw = (const float*)d_in[9];
  const float* pab = (const float*)d_in[10];
  const float* w1  = (const float*)d_in[11];
  const float* bu1 = (const float*)d_in[12];
  const float* ba1 = (const float*)d_in[13];
  const float* w2  = (const float*)d_in[14];
  const float* bu2 = (const float*)d_in[15];
  const float* ba2 = (const float*)d_in[16];
  const float* wc  = (const float*)d_in[17];
  const float* buc = (const float*)d_in[18];
  const float* bac = (const float*)d_in[19];
  float* out = (float*)d_out;

  float* ws    = (float*)d_ws;           // ~18.3 MB total scratch
  float* y     = ws;                     // 64*64*256      = 1,048,576
  float* pose0 = y + 1048576;            // 16384*128      = 2,097,152
  float* a0    = pose0 + 2097152;        // 16384*8        =   131,072
  float* pose1 = a0 + 131072;            // 3136*256       =   802,816
  float* a1    = pose1 + 802816;         // 3136*16        =    50,176
  float* pose2 = a1 + 50176;             // 1600*256       =   409,600
  float* a2    = pose2 + 409600;         // 1600*16        =    25,600

  // allow >64KB dynamic LDS for the fused caps kernel (class stage ~254KB)
  hipFuncSetAttribute((const void*)caps_layer_kernel,
                      hipFuncAttributeMaxDynamicSharedMemorySize, 262144);

  conv1_bn_relu<<<4096, 256, 0, stream>>>(x, c1w, c1b, bng, bnb, bnm, bnv, y);
  prim_pose_gemm<<<2048, 128, 0, stream>>>(y, ppw, ppb, pose0);
  prim_a_kernel<<<512, 256, 0, stream>>>(y, paw, pab, a0);

  // stage 1: Hin=16 Bin=8 K=3 s=2 -> 7x7x16, N=72, 784 blocks
  caps_layer_kernel<<<784, 128, caps_lds_bytes(72, 16, 4), stream>>>(
      pose0, a0, w1, bu1, ba1, pose1, a1,
      16, 16, 8, 3, 2, 16, 49, 7, 72, 4, 0);
  // stage 2: Hin=7 Bin=16 K=3 s=1 -> 5x5x16, N=144, 400 blocks
  caps_layer_kernel<<<400, 128, caps_lds_bytes(144, 16, 4), stream>>>(
      pose1, a1, w2, bu2, ba2, pose2, a2,
      7, 7, 16, 3, 1, 16, 25, 5, 144, 4, 0);
  // class caps: N=400, C=10 (MT=3), coord-add on votes, out = activations
  caps_layer_kernel<<<16, 128, caps_lds_bytes(400, 10, 3), stream>>>(
      pose2, a2, wc, buc, bac, nullptr, out,
      5, 5, 16, 1, 1, 10, 1, 1, 400, 3, 1);
}